// PairFeaturesLayer_23983097381473
// MI455X (gfx1250) — compile-verified
//
#include <hip/hip_runtime.h>

typedef __attribute__((ext_vector_type(2))) float v2f;
typedef __attribute__((ext_vector_type(4))) float v4f;
typedef __attribute__((ext_vector_type(8))) float v8f;

#define HDIM 128
#define NNODE 1024

// ---- CDNA5 async global->LDS path (guarded: falls back to reg staging) ----
#if defined(__has_builtin)
#if __has_builtin(__builtin_amdgcn_global_load_async_to_lds_b128)
#define USE_ASYNC_LDS 1
#endif
#endif
#ifndef USE_ASYNC_LDS
#define USE_ASYNC_LDS 0
#endif

#if USE_ASYNC_LDS
// Builtin parameter type per hipcc diagnostic:
//   'int __attribute__((vector_size(16))) __device__ *'  (AS1 for global)
typedef int b128_t __attribute__((vector_size(16)));
typedef __attribute__((address_space(1))) b128_t* gvec_t;
typedef __attribute__((address_space(3))) b128_t* lvec_t;

__device__ __forceinline__ gvec_t to_global(const void* p) {
  return (gvec_t)(uintptr_t)p;  // generic global addr == AS1 addr
}
__device__ __forceinline__ lvec_t to_lds(const void* p) {
  // LDS aperture: generic addr[31:0] is the LDS byte address (ISA 10.2)
  return (lvec_t)(uint32_t)(uintptr_t)p;
}
__device__ __forceinline__ void wait_async0() {
#if __has_builtin(__builtin_amdgcn_s_wait_asynccnt)
  __builtin_amdgcn_s_wait_asynccnt(0);
#else
  asm volatile("s_wait_asynccnt 0x0" ::: "memory");
#endif
}
#endif

// ---------------------------------------------------------------------------
// Kernel 1: f_i = X @ Wi^T, f_j = X @ Wj^T   (tiny: 2 * 1024*128*128 MACs)
// ---------------------------------------------------------------------------
__global__ __launch_bounds__(256) void fi_fj_kernel(
    const float* __restrict__ X, const float* __restrict__ Wi,
    const float* __restrict__ Wj, float* __restrict__ fi,
    float* __restrict__ fj) {
  int idx = blockIdx.x * blockDim.x + threadIdx.x;  // [0, 1024*128)
  int r = idx >> 7;
  int c = idx & 127;
  const float* xr = X + r * HDIM;
  const float* wi = Wi + c * HDIM;
  const float* wj = Wj + c * HDIM;
  float si = 0.f, sj = 0.f;
#pragma unroll 8
  for (int k = 0; k < HDIM; ++k) {
    float x = xr[k];
    si = fmaf(x, wi[k], si);
    sj = fmaf(x, wj[k], sj);
  }
  fi[idx] = si;
  fj[idx] = sj;
}

// ---------------------------------------------------------------------------
// Kernel 2: dots = f_i @ f_j^T  (1024x1024, K=128) — WMMA f32 16x16x4
// One wave per 16x16 output tile; 8 waves/block -> 512 blocks.
// ---------------------------------------------------------------------------
__global__ __launch_bounds__(256) void dots_kernel(
    const float* __restrict__ fi, const float* __restrict__ fj,
    float* __restrict__ dots) {
  int lane = threadIdx.x & 31;
  int w = threadIdx.x >> 5;
  int id = blockIdx.x * 8 + w;  // tile id in [0, 4096)
  int ti = id >> 6;
  int tj = id & 63;
  int half = lane >> 4;
  int ln = lane & 15;

  const float* arow = fi + (size_t)(ti * 16 + ln) * HDIM + 2 * half;
  const float* brow = fj + (size_t)(tj * 16 + ln) * HDIM + 2 * half;

  v8f acc = {0.f, 0.f, 0.f, 0.f, 0.f, 0.f, 0.f, 0.f};
#pragma unroll
  for (int s = 0; s < 32; ++s) {
    v2f af = *(const v2f*)(arow + 4 * s);
    v2f bf = *(const v2f*)(brow + 4 * s);
    acc = __builtin_amdgcn_wmma_f32_16x16x4_f32(false, af, false, bf, (short)0,
                                                acc, false, false);
  }

  float* o = dots + (size_t)(ti * 16 + 8 * half) * NNODE + tj * 16 + ln;
#pragma unroll
  for (int v = 0; v < 8; ++v) o[(size_t)v * NNODE] = acc[v];
}

// ---------------------------------------------------------------------------
// Kernel 3: out = relu(edge_attr @ We^T + be + dots[i,j])  [1M x 128]
// 8 waves/block; wave w owns output columns [16w,16w+16). We^T fragments are
// VGPR-resident (loaded once per block). A tiles double-buffered in LDS:
//   async path : global_load_async_to_lds_b128 + s_wait_asynccnt (ASYNCcnt)
//   fallback   : global->VGPR issued before compute, ds_store after
// ---------------------------------------------------------------------------
__global__ __launch_bounds__(256) void fused_kernel(
    const float* __restrict__ EA, const float* __restrict__ We,
    const float* __restrict__ be, const float* __restrict__ dots,
    float* __restrict__ out) {
  __shared__ __align__(16) float Atile[2][16 * HDIM];  // 2 x 8 KB

  const int tid = threadIdx.x;
  const int lane = tid & 31;
  const int w = tid >> 5;  // wave 0..7
  const int half = lane >> 4;
  const int ln = lane & 15;
  const int n0 = w * 16;

  // B fragments: b[s][p] = We[(n0+ln)][4s + p + 2*half]   (B[k,n] = We[n,k])
  const float* wrow = We + (size_t)(n0 + ln) * HDIM + 2 * half;
  v2f bfrag[32];
#pragma unroll
  for (int s = 0; s < 32; ++s) bfrag[s] = *(const v2f*)(wrow + 4 * s);
  float bias = be[n0 + ln];

  const int tiles = (NNODE * NNODE) / 16;  // 65536
  const int stride = gridDim.x;

  // ---- prologue: stage first tile into buffer 0 ----
  int t = blockIdx.x;
#if USE_ASYNC_LDS
  {
    gvec_t g = to_global(EA + (size_t)t * (16 * HDIM));
    lvec_t l = to_lds(&Atile[0][0]);
    __builtin_amdgcn_global_load_async_to_lds_b128(g + tid, l + tid, 0, 0);
    __builtin_amdgcn_global_load_async_to_lds_b128(g + 256 + tid, l + 256 + tid,
                                                   0, 0);
    wait_async0();
  }
#else
  {
    const v4f* src = (const v4f*)(EA + (size_t)t * (16 * HDIM));
    v4f* dst = (v4f*)Atile[0];
    dst[tid] = src[tid];
    dst[tid + 256] = src[tid + 256];
  }
#endif
  __syncthreads();

  int cur = 0;
  for (; t < tiles; t += stride) {
    const int tn = t + stride;  // uniform across block
    const bool have_next = tn < tiles;

#if USE_ASYNC_LDS
    if (have_next) {  // kick off next tile's DMA before compute
      gvec_t g = to_global(EA + (size_t)tn * (16 * HDIM));
      lvec_t l = to_lds(&Atile[cur ^ 1][0]);
      __builtin_amdgcn_global_load_async_to_lds_b128(g + tid, l + tid, 0, 0);
      __builtin_amdgcn_global_load_async_to_lds_b128(g + 256 + tid,
                                                     l + 256 + tid, 0, 0);
    }
#else
    v4f r0, r1;
    if (have_next) {  // issue global loads early; store to LDS after compute
      const v4f* src = (const v4f*)(EA + (size_t)tn * (16 * HDIM));
      r0 = src[tid];
      r1 = src[tid + 256];
    }
#endif

    // ---- accumulator init: bias + dots broadcast ----
    int r0i = t * 16;
    int i = r0i >> 10;
    int j0 = r0i & 1023;
    const float* dr = dots + (size_t)i * NNODE + j0 + 8 * half;
    v4f d0 = *(const v4f*)dr;
    v4f d1 = *(const v4f*)(dr + 4);
    v8f acc;
    acc[0] = bias + d0.x; acc[1] = bias + d0.y;
    acc[2] = bias + d0.z; acc[3] = bias + d0.w;
    acc[4] = bias + d1.x; acc[5] = bias + d1.y;
    acc[6] = bias + d1.z; acc[7] = bias + d1.w;

    // ---- 32 x WMMA f32 16x16x4 over K=128 from LDS buffer `cur` ----
    const float* arow = Atile[cur] + ln * HDIM + 2 * half;
#pragma unroll
    for (int s = 0; s < 32; ++s) {
      v2f af = *(const v2f*)(arow + 4 * s);
      acc = __builtin_amdgcn_wmma_f32_16x16x4_f32(false, af, false, bfrag[s],
                                                  (short)0, acc, false, false);
    }

    // ---- ReLU + store: lane (v,l) -> row v+8*half, col n0+ln ----
    float* orow = out + (size_t)t * (16 * HDIM) + (size_t)(8 * half) * HDIM +
                  n0 + ln;
#pragma unroll
    for (int v = 0; v < 8; ++v) orow[(size_t)v * HDIM] = fmaxf(acc[v], 0.0f);

#if USE_ASYNC_LDS
    wait_async0();  // next tile's DMA complete (this wave)
#else
    if (have_next) {
      v4f* dst = (v4f*)Atile[cur ^ 1];
      dst[tid] = r0;
      dst[tid + 256] = r1;
    }
#endif
    __syncthreads();  // cross-wave visibility of buffer cur^1
    cur ^= 1;
  }
}

// ---------------------------------------------------------------------------
extern "C" void kernel_launch(void* const* d_in, const int* in_sizes, int n_in,
                              void* d_out, int out_size, void* d_ws,
                              size_t ws_size, hipStream_t stream) {
  const float* X  = (const float*)d_in[0];  // node_feats [1024,128]
  const float* EA = (const float*)d_in[1];  // edge_attr  [1M,128]
  const float* We = (const float*)d_in[2];  // [128,128]
  const float* be = (const float*)d_in[3];  // [128]
  const float* Wi = (const float*)d_in[4];
  const float* Wj = (const float*)d_in[5];
  float* out = (float*)d_out;

  float* fi   = (float*)d_ws;               // 1024*128
  float* fj   = fi + NNODE * HDIM;          // 1024*128
  float* dots = fj + NNODE * HDIM;          // 1024*1024

  hipLaunchKernelGGL(fi_fj_kernel, dim3((NNODE * HDIM) / 256), dim3(256), 0,
                     stream, X, Wi, Wj, fi, fj);
  hipLaunchKernelGGL(dots_kernel, dim3(512), dim3(256), 0, stream, fi, fj,
                     dots);
  hipLaunchKernelGGL(fused_kernel, dim3(1024), dim3(256), 0, stream, EA, We,
                     be, dots, out);
}